// Model_45947560133156
// MI455X (gfx1250) — compile-verified
//
#include <hip/hip_runtime.h>
#include <hip/hip_bf16.h>

#define N_ALL 50000
#define N1    16000
#define N2    4096
#define E0    30000
#define E1    10000
#define SLOPE 0.01f
#define BN_EPS 1e-5f

typedef float v2f __attribute__((ext_vector_type(2)));
typedef float v4f __attribute__((ext_vector_type(4)));
typedef float v8f __attribute__((ext_vector_type(8)));

__device__ static inline float lrelu(float x) { return x > 0.f ? x : SLOPE * x; }

// D = A(16x4) * B(4x16) + C, full fp32 WMMA (wave32)
__device__ static inline v8f wmma4(v2f a, v2f b, v8f c) {
  return __builtin_amdgcn_wmma_f32_16x16x4_f32(false, a, false, b, (short)0, c, false, false);
}

// A fragment: row-major [16 x K] source in LDS; element A[lane&15][k0 + 2*(lane>>4) + r]
__device__ static inline v2f ldA(const float* base, int ld, int k0) {
  int m  = threadIdx.x & 15;
  int kk = k0 + 2 * ((threadIdx.x >> 4) & 1);
  v2f a;
  a.x = base[m * ld + kk];
  a.y = base[m * ld + kk + 1];
  return a;
}

// ---------------- weight repack: W[K][N] row-major -> fragment-native quads ----------------
// Wq flat index 8*(q*N+n) + 4*h + j  holds  W[8q + 2h + 4*(j>>1) + (j&1)][n].
// One b128 load per lane then feeds two WMMAs (k0=8q and k0=8q+4); 32 lanes cover a
// fully contiguous 512B block.
__global__ void repack_kernel(const float* __restrict__ W, int K, int N, float* __restrict__ Wq) {
  int idx = blockIdx.x * blockDim.x + threadIdx.x;
  if (idx >= K * N) return;
  int j = idx & 3;
  int h = (idx >> 2) & 1;
  int rem = idx >> 3;
  int n = rem % N;
  int q = rem / N;
  int k = 8 * q + 2 * h + 4 * (j >> 1) + (j & 1);
  Wq[idx] = W[k * N + n];
}

// B fragment pair from repacked weights
__device__ static inline v4f ldBq(const float* __restrict__ Wq, int N, int q, int nabs) {
  int h = (threadIdx.x >> 4) & 1;
  return *(const v4f*)(Wq + (((size_t)q * N + nabs) * 2 + h) * 4);
}

// One 16-row MLP layer: dst16xN = act(A16xK @ W + b). A in LDS, Wq repacked global,
// dst in LDS. Single wave (blockDim=32). K must be a multiple of 8.
__device__ static inline void gemm_tile(const float* Alds, int lda, int K,
                                        const float* Wq, const float* b, int N,
                                        float* dst, int ldd, bool act) {
  int col16 = threadIdx.x & 15;
  int rbase = 8 * ((threadIdx.x >> 4) & 1);
  for (int n0 = 0; n0 < N; n0 += 16) {
    float bias = b[n0 + col16];
    v8f c;
#pragma unroll
    for (int v = 0; v < 8; ++v) c[v] = bias;   // bias per-column => same for all C rows
    for (int q = 0; q < (K >> 3); ++q) {
      v4f w  = ldBq(Wq, N, q, n0 + col16);
      v2f a0 = ldA(Alds, lda, 8 * q);
      v2f a1 = ldA(Alds, lda, 8 * q + 4);
      c = wmma4(a0, w.xy, c);
      c = wmma4(a1, w.zw, c);
    }
#pragma unroll
    for (int v = 0; v < 8; ++v) {
      float x = c[v];
      if (act) x = lrelu(x);
      dst[(rbase + v) * ldd + (n0 + col16)] = x;
    }
  }
  __syncthreads();  // single-wave WG: barrier is S_NOP; LDS is in-order
}

// ---------------- embedder: h[idx[r]] = lrelu-MLP(x[r]; 32->64->64->64) ----------------
__global__ void embed_kernel(const float* __restrict__ x, const int* __restrict__ idx, int M,
                             const float* W0, const float* b0,
                             const float* W1, const float* b1,
                             const float* W2, const float* b2,
                             float* __restrict__ hout) {
  __shared__ float xs[16 * 33];
  __shared__ float t1[16 * 65];
  __shared__ float t2[16 * 65];
  int tile = blockIdx.x * 16;
  int lane = threadIdx.x;
  for (int i = lane; i < 16 * 32; i += 32) {
    int r = i >> 5, c = i & 31;
    xs[r * 33 + c] = (tile + r < M) ? x[(tile + r) * 32 + c] : 0.f;
  }
  __syncthreads();
  gemm_tile(xs, 33, 32, W0, b0, 64, t1, 65, true);
  gemm_tile(t1, 65, 64, W1, b1, 64, t2, 65, true);
  gemm_tile(t2, 65, 64, W2, b2, 64, t1, 65, true);  // final_act = True for embedders
  for (int i = lane; i < 16 * 64; i += 32) {
    int r = i >> 6, c = i & 63;
    if (tile + r < M) hout[(size_t)idx[tile + r] * 64 + c] = t1[r * 65 + c];
  }
}

// ---------------- fused edge_nn + per-edge matvec + scatter-sum ----------------
__global__ void edge_msg_kernel(const float* __restrict__ ef, const int* __restrict__ src,
                                const int* __restrict__ dst, int E,
                                const float* __restrict__ hsrc,
                                const float* W0, const float* b0,
                                const float* W1, const float* b1,
                                const float* W2q, const float* b2,
                                float* __restrict__ S, float* __restrict__ cnt) {
  __shared__ float efs[16 * 17];
  __shared__ float t1[16 * 65];
  __shared__ float h2[16 * 65];
  __shared__ float hs[16 * 65];
  __shared__ int   srcs[16];
  __shared__ int   dsts[16];
  int tile = blockIdx.x * 16;
  int lane = threadIdx.x;

  if (lane < 16) {
    int e = tile + lane;
    srcs[lane] = (e < E) ? src[e] : 0;
    dsts[lane] = (e < E) ? dst[e] : -1;
  }
  for (int i = lane; i < 16 * 16; i += 32) {
    int r = i >> 4, c = i & 15;
    efs[r * 17 + c] = (tile + r < E) ? ef[(tile + r) * 16 + c] : 0.f;
  }
  __syncthreads();
  for (int i = lane; i < 16 * 64; i += 32) {
    int r = i >> 6, c = i & 63;
    hs[r * 65 + c] = (tile + r < E) ? hsrc[(size_t)srcs[r] * 64 + c] : 0.f;
  }
  __syncthreads();

  gemm_tile(efs, 17, 16, W0, b0, 64, t1, 65, true);
  gemm_tile(t1, 65, 64, W1, b1, 64, h2, 65, true);

  v2f Af[16];
#pragma unroll
  for (int k = 0; k < 16; ++k) Af[k] = ldA(h2, 65, 4 * k);

  v8f msg[4];
#pragma unroll
  for (int s4 = 0; s4 < 4; ++s4) msg[s4] = (v8f)0.f;

  int col16 = lane & 15;
  int rbase = 8 * ((lane >> 4) & 1);

  for (int i = 0; i < 64; ++i) {            // input channel of edge-conditioned weight
    float hv[8];
#pragma unroll
    for (int v = 0; v < 8; ++v) hv[v] = hs[(rbase + v) * 65 + i];
#pragma unroll
    for (int n0 = 0; n0 < 64; n0 += 16) {   // output channel slice
      int nabs = i * 64 + n0 + col16;
      float bias = b2[nabs];
      v8f c;
#pragma unroll
      for (int v = 0; v < 8; ++v) c[v] = bias;
#pragma unroll
      for (int q = 0; q < 8; ++q) {         // K = 64: 8 b128 loads feed 16 WMMAs
        v4f w = ldBq(W2q, 4096, q, nabs);
        c = wmma4(Af[2 * q], w.xy, c);
        c = wmma4(Af[2 * q + 1], w.zw, c);
      }
      int s4 = n0 >> 4;
#pragma unroll
      for (int v = 0; v < 8; ++v)
        msg[s4][v] += hv[v] * lrelu(c[v]);
    }
  }

#pragma unroll
  for (int s4 = 0; s4 < 4; ++s4) {
#pragma unroll
    for (int v = 0; v < 8; ++v) {
      int d = dsts[rbase + v];
      if (d >= 0) atomicAdd(&S[(size_t)d * 64 + s4 * 16 + col16], msg[s4][v]);
    }
  }
  if (lane < 16 && dsts[lane] >= 0) atomicAdd(&cnt[dsts[lane]], 1.0f);
}

// ---------------- BN statistics over lrelu(concat(agg, h_tgt)) ----------------
__global__ void bn_stats_kernel(const float* __restrict__ S, const float* __restrict__ cnt,
                                const float* __restrict__ htgt, int n_tgt, int rpb,
                                float* __restrict__ part) {
  int c  = threadIdx.x;               // 0..127 column
  int r0 = blockIdx.x * rpb;
  int r1 = r0 + rpb; if (r1 > n_tgt) r1 = n_tgt;
  float s = 0.f, ss = 0.f;
  for (int r = r0; r < r1; ++r) {
    float x;
    if (c < 64) x = S[(size_t)r * 64 + c] / fmaxf(cnt[r], 1.f);
    else        x = htgt[(size_t)r * 64 + (c - 64)];
    x = lrelu(x);
    s += x; ss += x * x;
  }
  part[blockIdx.x * 256 + c]       = s;
  part[blockIdx.x * 256 + 128 + c] = ss;
}

__global__ void bn_reduce_kernel(const float* __restrict__ part, int nblk, int n_tgt,
                                 const float* __restrict__ g, const float* __restrict__ b,
                                 float* __restrict__ scale, float* __restrict__ shift) {
  int c = threadIdx.x;
  float s = 0.f, ss = 0.f;
  for (int i = 0; i < nblk; ++i) { s += part[i * 256 + c]; ss += part[i * 256 + 128 + c]; }
  float m   = s / (float)n_tgt;
  float var = ss / (float)n_tgt - m * m;
  float sc  = g[c] * rsqrtf(var + BN_EPS);
  scale[c] = sc;
  shift[c] = b[c] - m * sc;
}

// ---------------- node_nn: BN applied on the fly -> 128->64->64->64 (no final act) ----------------
__global__ void node_mlp_kernel(const float* __restrict__ S, const float* __restrict__ cnt,
                                const float* __restrict__ htgt,
                                const float* __restrict__ scale, const float* __restrict__ shift,
                                int n_tgt,
                                const float* W0, const float* b0,
                                const float* W1, const float* b1,
                                const float* W2, const float* b2,
                                float* __restrict__ hout) {
  __shared__ float xb[16 * 129];
  __shared__ float t1[16 * 65];
  __shared__ float t2[16 * 65];
  int tile = blockIdx.x * 16;
  int lane = threadIdx.x;
  for (int i = lane; i < 16 * 128; i += 32) {
    int r = i >> 7, c = i & 127;
    int n = tile + r;
    float x = 0.f;
    if (n < n_tgt) {
      if (c < 64) x = S[(size_t)n * 64 + c] / fmaxf(cnt[n], 1.f);
      else        x = htgt[(size_t)n * 64 + (c - 64)];
      x = lrelu(x) * scale[c] + shift[c];
    }
    xb[r * 129 + c] = x;
  }
  __syncthreads();
  gemm_tile(xb, 129, 128, W0, b0, 64, t1, 65, true);
  gemm_tile(t1, 65, 64,  W1, b1, 64, t2, 65, true);
  gemm_tile(t2, 65, 64,  W2, b2, 64, t1, 65, false);
  for (int i = lane; i < 16 * 64; i += 32) {
    int r = i >> 6, c = i & 63;
    if (tile + r < n_tgt) hout[(size_t)(tile + r) * 64 + c] = t1[r * 65 + c];
  }
}

// ---------------- head: out = lrelu(h@W1+b1)@W2 + b2  (64->64->8) ----------------
__global__ void head_kernel(const float* __restrict__ hin, int n,
                            const float* W1q, const float* b1,
                            const float* W2, const float* b2,   // W2 raw (64x8), tiny
                            float* __restrict__ out) {
  __shared__ float xb[16 * 65];
  __shared__ float t1[16 * 65];
  int tile = blockIdx.x * 16;
  int lane = threadIdx.x;
  for (int i = lane; i < 16 * 64; i += 32) {
    int r = i >> 6, c = i & 63;
    xb[r * 65 + c] = (tile + r < n) ? hin[(size_t)(tile + r) * 64 + c] : 0.f;
  }
  __syncthreads();
  gemm_tile(xb, 65, 64, W1q, b1, 64, t1, 65, true);

  int col16 = lane & 15;
  int rbase = 8 * ((lane >> 4) & 1);
  v8f c = (v8f)0.f;
  for (int k0 = 0; k0 < 64; k0 += 4) {
    v2f a = ldA(t1, 65, k0);
    int kk = k0 + 2 * ((lane >> 4) & 1);
    v2f bb;
    bb.x = (col16 < 8) ? W2[kk * 8 + col16] : 0.f;
    bb.y = (col16 < 8) ? W2[(kk + 1) * 8 + col16] : 0.f;
    c = wmma4(a, bb, c);
  }
  if (col16 < 8) {
    float bias = b2[col16];
#pragma unroll
    for (int v = 0; v < 8; ++v) {
      int r = rbase + v;
      if (tile + r < n) out[(size_t)(tile + r) * 8 + col16] = c[v] + bias;
    }
  }
}

__global__ void zero_kernel(float* p, int n) {
  int i = blockIdx.x * blockDim.x + threadIdx.x;
  if (i < n) p[i] = 0.f;
}

struct MlpP { const float *W0, *b0, *W1, *b1, *W2, *b2; };
struct ConvP { const float *bng, *bnb; MlpP edge, node; };

extern "C" void kernel_launch(void* const* d_in, const int* in_sizes, int n_in,
                              void* d_out, int out_size, void* d_ws, size_t ws_size,
                              hipStream_t stream) {
  auto f  = [&](int i) { return (const float*)d_in[i]; };
  auto ii = [&](int i) { return (const int*)d_in[i]; };

  ConvP c0, c1; MlpP embt, embo;
  auto mlp6 = [&](int base) { return MlpP{f(base), f(base + 1), f(base + 2), f(base + 3), f(base + 4), f(base + 5)}; };

  if (in_sizes[0] == 128) {
    // jax-sorted dict order: conv0{bn_b,bn_g,edge,node}, conv1, emb_o, emb_t, lin1, lin2
    c0 = ConvP{f(1), f(0), mlp6(2), mlp6(8)};
    c1 = ConvP{f(15), f(14), mlp6(16), mlp6(22)};
    embo = mlp6(28); embt = mlp6(34);
  } else {
    // insertion order: emb_t, emb_o, conv0{edge,node,bn_g,bn_b}, conv1, lin1, lin2
    embt = mlp6(0); embo = mlp6(6);
    c0 = ConvP{f(24), f(25), mlp6(12), mlp6(18)};
    c1 = ConvP{f(38), f(39), mlp6(26), mlp6(32)};
  }
  const float *l1W = f(40), *l1b = f(41), *l2W = f(42), *l2b = f(43);
  const float* x_target = f(44);
  const float* x_other  = f(45);
  const int *id_t = ii(46), *id_o = ii(47);
  const int *src0 = ii(48), *dst0 = ii(49);
  const float* ef0 = f(50);
  const int *src1 = ii(51), *dst1 = ii(52);
  const float* ef1 = f(53);

  // workspace layout (all region sizes multiples of 4 floats -> 16B alignment preserved)
  float* ws = (float*)d_ws;
  float* h     = ws;        size_t off = (size_t)N_ALL * 64;
  float* S     = ws + off;  off += (size_t)N1 * 64;
  float* cnt   = ws + off;  off += N1;
  float* part  = ws + off;  off += 256 * 256;
  float* scale = ws + off;  off += 128;
  float* shift = ws + off;  off += 128;
  float* hn1   = ws + off;  off += (size_t)N1 * 64;
  float* hn2   = ws + off;  off += (size_t)N2 * 64;

  auto alloc = [&](size_t n) { float* p = ws + off; off += n; return p; };
  float* q_c0e0 = alloc(16 * 64);   float* q_c0e1 = alloc(64 * 64);  float* q_c0e2 = alloc((size_t)64 * 4096);
  float* q_c0n0 = alloc(128 * 64);  float* q_c0n1 = alloc(64 * 64);  float* q_c0n2 = alloc(64 * 64);
  float* q_c1e0 = alloc(16 * 64);   float* q_c1e1 = alloc(64 * 64);  float* q_c1e2 = alloc((size_t)64 * 4096);
  float* q_c1n0 = alloc(128 * 64);  float* q_c1n1 = alloc(64 * 64);  float* q_c1n2 = alloc(64 * 64);
  float* q_et0  = alloc(32 * 64);   float* q_et1  = alloc(64 * 64);  float* q_et2  = alloc(64 * 64);
  float* q_eo0  = alloc(32 * 64);   float* q_eo1  = alloc(64 * 64);  float* q_eo2  = alloc(64 * 64);
  float* q_l1   = alloc(64 * 64);

  struct Job { const float* W; int K, N; float* out; };
  const Job jobs[] = {
    {c0.edge.W0, 16, 64, q_c0e0}, {c0.edge.W1, 64, 64, q_c0e1}, {c0.edge.W2, 64, 4096, q_c0e2},
    {c0.node.W0, 128, 64, q_c0n0}, {c0.node.W1, 64, 64, q_c0n1}, {c0.node.W2, 64, 64, q_c0n2},
    {c1.edge.W0, 16, 64, q_c1e0}, {c1.edge.W1, 64, 64, q_c1e1}, {c1.edge.W2, 64, 4096, q_c1e2},
    {c1.node.W0, 128, 64, q_c1n0}, {c1.node.W1, 64, 64, q_c1n1}, {c1.node.W2, 64, 64, q_c1n2},
    {embt.W0, 32, 64, q_et0}, {embt.W1, 64, 64, q_et1}, {embt.W2, 64, 64, q_et2},
    {embo.W0, 32, 64, q_eo0}, {embo.W1, 64, 64, q_eo1}, {embo.W2, 64, 64, q_eo2},
    {l1W, 64, 64, q_l1},
  };
  for (const Job& j : jobs) {
    int tot = j.K * j.N;
    repack_kernel<<<(tot + 255) / 256, 256, 0, stream>>>(j.W, j.K, j.N, j.out);
  }

  const int MO = N_ALL - N2;  // 45904

  // ---- embed (all N_ALL rows covered by the two index sets) ----
  embed_kernel<<<(MO + 15) / 16, 32, 0, stream>>>(x_other, id_o, MO,
      q_eo0, embo.b0, q_eo1, embo.b1, q_eo2, embo.b2, h);
  embed_kernel<<<(N2 + 15) / 16, 32, 0, stream>>>(x_target, id_t, N2,
      q_et0, embt.b0, q_et1, embt.b1, q_et2, embt.b2, h);

  // ---- conv0: N_ALL sources -> N1 targets ----
  zero_kernel<<<(N1 * 64 + 255) / 256, 256, 0, stream>>>(S, N1 * 64);
  zero_kernel<<<(N1 + 255) / 256, 256, 0, stream>>>(cnt, N1);
  edge_msg_kernel<<<(E0 + 15) / 16, 32, 0, stream>>>(ef0, src0, dst0, E0, h,
      q_c0e0, c0.edge.b0, q_c0e1, c0.edge.b1, q_c0e2, c0.edge.b2, S, cnt);
  int nblk0 = (N1 + 63) / 64;
  bn_stats_kernel<<<nblk0, 128, 0, stream>>>(S, cnt, h, N1, 64, part);
  bn_reduce_kernel<<<1, 128, 0, stream>>>(part, nblk0, N1, c0.bng, c0.bnb, scale, shift);
  node_mlp_kernel<<<(N1 + 15) / 16, 32, 0, stream>>>(S, cnt, h, scale, shift, N1,
      q_c0n0, c0.node.b0, q_c0n1, c0.node.b1, q_c0n2, c0.node.b2, hn1);

  // ---- conv1: N1 sources -> N2 targets ----
  zero_kernel<<<(N2 * 64 + 255) / 256, 256, 0, stream>>>(S, N2 * 64);
  zero_kernel<<<(N2 + 255) / 256, 256, 0, stream>>>(cnt, N2);
  edge_msg_kernel<<<(E1 + 15) / 16, 32, 0, stream>>>(ef1, src1, dst1, E1, hn1,
      q_c1e0, c1.edge.b0, q_c1e1, c1.edge.b1, q_c1e2, c1.edge.b2, S, cnt);
  int nblk1 = (N2 + 63) / 64;
  bn_stats_kernel<<<nblk1, 128, 0, stream>>>(S, cnt, hn1, N2, 64, part);
  bn_reduce_kernel<<<1, 128, 0, stream>>>(part, nblk1, N2, c1.bng, c1.bnb, scale, shift);
  node_mlp_kernel<<<(N2 + 15) / 16, 32, 0, stream>>>(S, cnt, hn1, scale, shift, N2,
      q_c1n0, c1.node.b0, q_c1n1, c1.node.b1, q_c1n2, c1.node.b2, hn2);

  // ---- head ----
  head_kernel<<<(N2 + 15) / 16, 32, 0, stream>>>(hn2, N2, q_l1, l1b, l2W, l2b, (float*)d_out);
}